// Conv_62861141344336
// MI455X (gfx1250) — compile-verified
//
#include <hip/hip_runtime.h>

#define NNODES 10000
#define NEDGES 50000
#define DIM 32
#define NH 4

typedef float v2f __attribute__((ext_vector_type(2)));
typedef float v8f __attribute__((ext_vector_type(8)));

// ---- order-preserving float <-> u32 encoding for atomic max ----
__device__ __forceinline__ unsigned enc_f(float x) {
  unsigned u = __float_as_uint(x);
  return (u & 0x80000000u) ? ~u : (u | 0x80000000u);
}
__device__ __forceinline__ float dec_f(unsigned e) {
  return (e & 0x80000000u) ? __uint_as_float(e & 0x7FFFFFFFu)
                           : __uint_as_float(~e);
}

// ---- 32x32 f32 matvec on the WMMA pipe: acc += W(32x32) * x ----
// One wave per matvec. W row-major, contiguous 4KB block.
// A frag (16x4 f32): lanes 0-15 k={0,1}, lanes 16-31 k={2,3}.
// B frag (4x16): same k-grouping, value broadcast across all 16 columns.
__device__ __forceinline__ void matvec32_wmma(const float* __restrict__ W,
                                              float xreg, int lane,
                                              v8f& acc_lo, v8f& acc_hi) {
  const int r  = lane & 15;
  const int cb = (lane >> 4) << 1;          // 0 for lanes<16, 2 for lanes>=16
  const float* rowlo = W + r * DIM;         // outputs 0..15
  const float* rowhi = W + (r + 16) * DIM;  // outputs 16..31
#pragma unroll
  for (int c = 0; c < 8; ++c) {
    const int col = 4 * c + cb;
    v2f a_lo = *(const v2f*)(rowlo + col);
    v2f a_hi = *(const v2f*)(rowhi + col);
    v2f b;
    b.x = __shfl(xreg, col, 32);
    b.y = __shfl(xreg, col + 1, 32);
    acc_lo = __builtin_amdgcn_wmma_f32_16x16x4_f32(false, a_lo, false, b,
                                                   (short)0, acc_lo, false, false);
    acc_hi = __builtin_amdgcn_wmma_f32_16x16x4_f32(false, a_hi, false, b,
                                                   (short)0, acc_hi, false, false);
  }
}

// ---- init workspace (maxenc=0 encodes -inf, denom=0, agg=0) ----
__global__ __launch_bounds__(256) void init_kernel(unsigned* maxenc,
                                                   float* denom, float* agg) {
  int i = blockIdx.x * blockDim.x + threadIdx.x;
  if (i < NNODES * NH) { maxenc[i] = 0u; denom[i] = 0.0f; }
  if (i < NNODES * DIM) agg[i] = 0.0f;
}

// ---- per-edge: K/V hypernetwork matvecs (WMMA), logits, running max ----
__global__ __launch_bounds__(256) void edge_kernel(
    const float* __restrict__ in_feat, const int* __restrict__ src,
    const int* __restrict__ dst,
    const float* __restrict__ skw, const float* __restrict__ dkw,
    const float* __restrict__ skb, const float* __restrict__ dkb,
    const float* __restrict__ svw, const float* __restrict__ dvw,
    const float* __restrict__ svb, const float* __restrict__ dvb,
    const float* __restrict__ query,
    float* __restrict__ key_out, float* __restrict__ val_out,
    float* __restrict__ logits, unsigned* __restrict__ maxenc) {
  __shared__ float sbK[8][DIM];
  __shared__ float sbV[8][DIM];
  const int lane = threadIdx.x & 31;
  const int wv   = threadIdx.x >> 5;
  const int e    = blockIdx.x * 8 + wv;     // grid sized exactly: no guard
  const int s = src[e];
  const int d = dst[e];
  const float u  = in_feat[s * DIM + lane];
  const float vf = in_feat[d * DIM + lane];
  const size_t wo = (size_t)e * (DIM * DIM);

  v8f kl = {}, kh = {}, vl = {}, vh = {};
  matvec32_wmma(skw + wo, u,  lane, kl, kh);
  matvec32_wmma(dkw + wo, vf, lane, kl, kh);
  matvec32_wmma(svw + wo, u,  lane, vl, vh);
  matvec32_wmma(dvw + wo, vf, lane, vl, vh);

  // D layout: vgpr r / lanes 0-15 -> M=r ; lanes 16-31 -> M=r+8 (cols identical)
  if (lane == 0) {
#pragma unroll
    for (int r = 0; r < 8; ++r) {
      sbK[wv][r] = kl[r]; sbK[wv][16 + r] = kh[r];
      sbV[wv][r] = vl[r]; sbV[wv][16 + r] = vh[r];
    }
  } else if (lane == 16) {
#pragma unroll
    for (int r = 0; r < 8; ++r) {
      sbK[wv][8 + r] = kl[r]; sbK[wv][24 + r] = kh[r];
      sbV[wv][8 + r] = vl[r]; sbV[wv][24 + r] = vh[r];
    }
  }
  __syncthreads();

  float K = sbK[wv][lane] + skb[e * DIM + lane] + dkb[e * DIM + lane];
  float V = sbV[wv][lane] + svb[e * DIM + lane] + dvb[e * DIM + lane];
  key_out[e * DIM + lane] = K;
  val_out[e * DIM + lane] = V;

  // logits[e,h] = sum_{hd} K[h,hd] * query[d,h,hd]  (8-lane groups)
  float p = K * query[d * DIM + lane];
  p += __shfl_xor(p, 4, 8);
  p += __shfl_xor(p, 2, 8);
  p += __shfl_xor(p, 1, 8);
  if ((lane & 7) == 0) {
    int h = lane >> 3;
    logits[e * NH + h] = p;
    atomicMax(&maxenc[d * NH + h], enc_f(p));
  }
}

// ---- softmax numerator + denominator ----
__global__ __launch_bounds__(256) void softmax_num_kernel(
    const int* __restrict__ dst, const float* __restrict__ logits,
    const unsigned* __restrict__ maxenc, float* __restrict__ exbuf,
    float* __restrict__ denom) {
  int i = blockIdx.x * blockDim.x + threadIdx.x;
  if (i >= NEDGES * NH) return;
  int e = i >> 2;
  int h = i & 3;
  int d = dst[e];
  float m  = dec_f(maxenc[d * NH + h]);
  float ex = __expf(logits[i] - m);
  exbuf[i] = ex;
  atomicAdd(&denom[d * NH + h], ex);
}

// ---- attn = ex/denom ; scatter agg[dst] += V * attn ----
__global__ __launch_bounds__(256) void attn_agg_kernel(
    const int* __restrict__ dst, const float* __restrict__ exbuf,
    const float* __restrict__ denom, const float* __restrict__ val_out,
    float* __restrict__ attn_out, float* __restrict__ agg) {
  int tid = blockIdx.x * blockDim.x + threadIdx.x;  // E*32 exact
  int e = tid >> 5;
  int j = tid & 31;
  int d = dst[e];
  int h = j >> 3;
  float a = exbuf[e * NH + h] / denom[d * NH + h];
  if ((j & 7) == 0) attn_out[e * NH + h] = a;
  atomicAdd(&agg[d * DIM + j], val_out[e * DIM + j] * a);
}

// ---- per-node dynamic linear (WMMA) + relu + residual + layernorm ----
__global__ __launch_bounds__(256) void node_kernel(
    const float* __restrict__ in_feat, const float* __restrict__ agg,
    const float* __restrict__ node_w, const float* __restrict__ node_b,
    const float* __restrict__ ln_w, const float* __restrict__ ln_b,
    float* __restrict__ out) {
  __shared__ float sbO[8][DIM];
  const int lane = threadIdx.x & 31;
  const int wv   = threadIdx.x >> 5;
  const int n    = blockIdx.x * 8 + wv;     // grid exact
  float x = agg[n * DIM + lane];
  v8f ol = {}, oh = {};
  matvec32_wmma(node_w + (size_t)n * (DIM * DIM), x, lane, ol, oh);
  if (lane == 0) {
#pragma unroll
    for (int r = 0; r < 8; ++r) { sbO[wv][r] = ol[r]; sbO[wv][16 + r] = oh[r]; }
  } else if (lane == 16) {
#pragma unroll
    for (int r = 0; r < 8; ++r) { sbO[wv][8 + r] = ol[r]; sbO[wv][24 + r] = oh[r]; }
  }
  __syncthreads();
  float o = sbO[wv][lane] + node_b[n * DIM + lane];
  o = fmaxf(o, 0.0f);
  float r = in_feat[n * DIM + lane] + o;
  float sum = r;
  sum += __shfl_xor(sum, 16, 32);
  sum += __shfl_xor(sum, 8, 32);
  sum += __shfl_xor(sum, 4, 32);
  sum += __shfl_xor(sum, 2, 32);
  sum += __shfl_xor(sum, 1, 32);
  float mu = sum * (1.0f / 32.0f);
  float df = r - mu;
  float v = df * df;
  v += __shfl_xor(v, 16, 32);
  v += __shfl_xor(v, 8, 32);
  v += __shfl_xor(v, 4, 32);
  v += __shfl_xor(v, 2, 32);
  v += __shfl_xor(v, 1, 32);
  float var = v * (1.0f / 32.0f);
  out[n * DIM + lane] = df * rsqrtf(var + 1e-5f) * ln_w[lane] + ln_b[lane];
}

extern "C" void kernel_launch(void* const* d_in, const int* in_sizes, int n_in,
                              void* d_out, int out_size, void* d_ws, size_t ws_size,
                              hipStream_t stream) {
  const float* in_feat = (const float*)d_in[0];
  const int*   src     = (const int*)d_in[1];
  const int*   dst     = (const int*)d_in[2];
  const float* skw     = (const float*)d_in[3];
  const float* dkw     = (const float*)d_in[4];
  const float* skb     = (const float*)d_in[5];
  const float* dkb     = (const float*)d_in[6];
  const float* svw     = (const float*)d_in[7];
  const float* dvw     = (const float*)d_in[8];
  const float* svb     = (const float*)d_in[9];
  const float* dvb     = (const float*)d_in[10];
  const float* query   = (const float*)d_in[11];
  const float* node_w  = (const float*)d_in[12];
  const float* node_b  = (const float*)d_in[13];
  const float* ln_w    = (const float*)d_in[14];
  const float* ln_b    = (const float*)d_in[15];

  // d_out: out (N*32) | key_feat (E*32) | value_feat (E*32) | attn (E*4)
  float* out_p  = (float*)d_out;
  float* key_p  = out_p + (size_t)NNODES * DIM;
  float* val_p  = key_p + (size_t)NEDGES * DIM;
  float* attn_p = val_p + (size_t)NEDGES * DIM;

  // workspace: logits | exbuf | maxenc | denom | agg   (3.2 MB total)
  float*    logits = (float*)d_ws;
  float*    exbuf  = logits + (size_t)NEDGES * NH;
  unsigned* maxenc = (unsigned*)(exbuf + (size_t)NEDGES * NH);
  float*    denom  = (float*)(maxenc + (size_t)NNODES * NH);
  float*    agg    = denom + (size_t)NNODES * NH;

  hipLaunchKernelGGL(init_kernel, dim3((NNODES * DIM + 255) / 256), dim3(256), 0,
                     stream, maxenc, denom, agg);
  hipLaunchKernelGGL(edge_kernel, dim3(NEDGES / 8), dim3(256), 0, stream,
                     in_feat, src, dst, skw, dkw, skb, dkb, svw, dvw, svb, dvb,
                     query, key_p, val_p, logits, maxenc);
  hipLaunchKernelGGL(softmax_num_kernel, dim3((NEDGES * NH + 255) / 256),
                     dim3(256), 0, stream, dst, logits, maxenc, exbuf, denom);
  hipLaunchKernelGGL(attn_agg_kernel, dim3(NEDGES * DIM / 256), dim3(256), 0,
                     stream, dst, exbuf, denom, val_p, attn_p, agg);
  hipLaunchKernelGGL(node_kernel, dim3(NNODES / 8), dim3(256), 0, stream,
                     in_feat, agg, node_w, node_b, ln_w, ln_b, out_p);
}